// MeshRenderer_1357209666431
// MI455X (gfx1250) — compile-verified
//
#include <hip/hip_runtime.h>
#include <hip/hip_bf16.h>

// Problem constants (from reference)
#define BS   4
#define NV   50000
#define NTRI 100000
#define HH   512
#define WW   512
#define HW   (HH * WW)          // 262144
#define NPIX (BS * HW)          // 1048576

typedef __attribute__((ext_vector_type(2))) float v2f;
typedef __attribute__((ext_vector_type(8))) float v8f;

// ---------------------------------------------------------------------------
// Kernel 1: zero the two per-vertex normal accumulators (world + transformed)
// ---------------------------------------------------------------------------
__global__ void mr_zero_kernel(float* __restrict__ acc, int n) {
    int i = blockIdx.x * blockDim.x + threadIdx.x;
    if (i < n) acc[i] = 0.0f;
}

// ---------------------------------------------------------------------------
// Kernel 2: per-face normals, scatter-add into per-vertex accumulators.
// Note: the z+=10 shift cancels in the edge differences, so the transformed
// face normal is computed directly from vertices_image_normalised.
// ---------------------------------------------------------------------------
__global__ void mr_face_kernel(const float* __restrict__ vw,   // (BS,NV,3)
                               const float* __restrict__ vi,   // (BS,NV,3)
                               const int*   __restrict__ tf,   // (BS,NTRI,3)
                               float* __restrict__ accW,       // (BS,NV,3)
                               float* __restrict__ accT) {     // (BS,NV,3)
    int F = blockIdx.x * blockDim.x + threadIdx.x;
    if (F >= BS * NTRI) return;
    int b = F / NTRI;
    const int* t = tf + (size_t)F * 3;
    int i0 = t[0], i1 = t[1], i2 = t[2];
    size_t base = (size_t)b * NV;

    // world-space face normal
    const float* p0 = vw + (base + i0) * 3;
    const float* p1 = vw + (base + i1) * 3;
    const float* p2 = vw + (base + i2) * 3;
    float e1x = p1[0] - p0[0], e1y = p1[1] - p0[1], e1z = p1[2] - p0[2];
    float e2x = p2[0] - p0[0], e2y = p2[1] - p0[1], e2z = p2[2] - p0[2];
    float wx = e1y * e2z - e1z * e2y;
    float wy = e1z * e2x - e1x * e2z;
    float wz = e1x * e2y - e1y * e2x;

    // transformed-space face normal (z-shift cancels)
    const float* q0 = vi + (base + i0) * 3;
    const float* q1 = vi + (base + i1) * 3;
    const float* q2 = vi + (base + i2) * 3;
    float f1x = q1[0] - q0[0], f1y = q1[1] - q0[1], f1z = q1[2] - q0[2];
    float f2x = q2[0] - q0[0], f2y = q2[1] - q0[1], f2z = q2[2] - q0[2];
    float tx = f1y * f2z - f1z * f2y;
    float ty = f1z * f2x - f1x * f2z;
    float tz = f1x * f2y - f1y * f2x;

    float* aw0 = accW + (base + i0) * 3;
    float* aw1 = accW + (base + i1) * 3;
    float* aw2 = accW + (base + i2) * 3;
    atomicAdd(aw0 + 0, wx); atomicAdd(aw0 + 1, wy); atomicAdd(aw0 + 2, wz);
    atomicAdd(aw1 + 0, wx); atomicAdd(aw1 + 1, wy); atomicAdd(aw1 + 2, wz);
    atomicAdd(aw2 + 0, wx); atomicAdd(aw2 + 1, wy); atomicAdd(aw2 + 2, wz);

    float* at0 = accT + (base + i0) * 3;
    float* at1 = accT + (base + i1) * 3;
    float* at2 = accT + (base + i2) * 3;
    atomicAdd(at0 + 0, tx); atomicAdd(at0 + 1, ty); atomicAdd(at0 + 2, tz);
    atomicAdd(at1 + 0, tx); atomicAdd(at1 + 1, ty); atomicAdd(at1 + 2, tz);
    atomicAdd(at2 + 0, tx); atomicAdd(at2 + 1, ty); atomicAdd(at2 + 2, tz);
}

// ---------------------------------------------------------------------------
// Kernel 3: normalize accumulators, pack (nWx, nWy, nWz, nTz) per vertex.
// ---------------------------------------------------------------------------
__global__ void mr_normalize_kernel(const float* __restrict__ accW,
                                    const float* __restrict__ accT,
                                    float4* __restrict__ packed) {
    int j = blockIdx.x * blockDim.x + threadIdx.x;
    if (j >= BS * NV) return;
    float wx = accW[(size_t)j * 3 + 0];
    float wy = accW[(size_t)j * 3 + 1];
    float wz = accW[(size_t)j * 3 + 2];
    float wl = sqrtf(wx * wx + wy * wy + wz * wz);
    float wi = 1.0f / fmaxf(wl, 1e-12f);
    float tx = accT[(size_t)j * 3 + 0];
    float ty = accT[(size_t)j * 3 + 1];
    float tz = accT[(size_t)j * 3 + 2];
    float tl = sqrtf(tx * tx + ty * ty + tz * tz);
    float ti = 1.0f / fmaxf(tl, 1e-12f);
    float4 o;
    o.x = wx * wi; o.y = wy * wi; o.z = wz * wi; o.w = tz * ti;
    packed[j] = o;
}

// ---------------------------------------------------------------------------
// Kernel 4: per-pixel gather + barycentric interp + WMMA lighting + composite.
// Each wave handles 16 pixels per WMMA (M = 16 lights [5 real], K = 4
// [3 normal components], N = 16 pixels). Lanes 0-15 carry (Nx,Ny) for K=0,1;
// lanes 16-31 carry (Nz,0) for K=2,3 — per the f32 A/B VGPR striping. The D
// layout leaves all 5 ndl values for pixel L in lane L's VGPRs 0..4, so the
// clip/mean reduction is lane-local. Total pixels divisible by 16 -> EXEC is
// always all-ones at the WMMA.
// ---------------------------------------------------------------------------
__global__ __launch_bounds__(256) void mr_pixel_kernel(
        const int*    __restrict__ p2f,    // (BS,H,W,1) flat
        const float*  __restrict__ bary,   // (BS,H,W,1,3) flat
        const int*    __restrict__ tf,     // (BS,NTRI,3)
        const float4* __restrict__ packed, // (BS,NV) {nWx,nWy,nWz,nTz}
        const float*  __restrict__ images, // (BS,3,H,W)
        float*        __restrict__ out) {  // (BS,3,H,W)
    const int lane = threadIdx.x & 31;
    const int m    = lane & 15;
    const bool hi  = (threadIdx.x & 16) != 0;
    const int wave = threadIdx.x >> 5;
    const long group = (long)blockIdx.x * 8 + wave;   // 8 waves per block
    const long p = group * 16 + m;                    // pixel this lane serves

    int f = p2f[p];
    bool mask = (f < 0);
    float nx = 0.0f, ny = 0.0f, nz = 0.0f, tz = 0.0f, bsum = 0.0f;
    if (!mask) {
        int fb  = f / NTRI;
        int tri = f - fb * NTRI;
        const int* t = tf + ((size_t)fb * NTRI + (size_t)tri) * 3;
        int i0 = t[0], i1 = t[1], i2 = t[2];
        float b0 = bary[p * 3 + 0];
        float b1 = bary[p * 3 + 1];
        float b2 = bary[p * 3 + 2];
        const float4* pk = packed + (size_t)fb * NV;
        float4 n0 = pk[i0];
        float4 n1 = pk[i1];
        float4 n2 = pk[i2];
        nx = b0 * n0.x + b1 * n1.x + b2 * n2.x;
        ny = b0 * n0.y + b1 * n1.y + b2 * n2.y;
        nz = b0 * n0.z + b1 * n1.z + b2 * n2.z;
        tz = b0 * n0.w + b1 * n1.w + b2 * n2.w;
        bsum = b0 + b1 + b2;
    }

    // normalized light dirs: rows 0-3 = (+-s,+-s,s), row 4 = (0,0,1), rest 0
    const float S = 0.57735026918962576f; // 1/sqrt(3)
    float dx = 0.0f, dy = 0.0f, dz = 0.0f;
    if (m < 4)       { dx = (m & 1) ? S : -S; dy = (m < 2) ? S : -S; dz = S; }
    else if (m == 4) { dx = 0.0f; dy = 0.0f; dz = 1.0f; }

    v2f a, b;
    if (!hi) { a.x = dx;   a.y = dy;   b.x = nx; b.y = ny; }  // K=0,1
    else     { a.x = dz;   a.y = 0.0f; b.x = nz; b.y = 0.0f; }// K=2,3
    v8f c = {};
    // D(light m, pixel n) = sum_k dirs(m,k) * N(k,n)
    v8f d = __builtin_amdgcn_wmma_f32_16x16x4_f32(
        /*neg_a=*/false, a, /*neg_b=*/false, b,
        /*c_mod=*/(short)0, c, /*reuse_a=*/false, /*reuse_b=*/false);

    if (!hi) {
        float s = 0.0f;
#pragma unroll
        for (int r = 0; r < 5; ++r)
            s += fminf(fmaxf(d[r], 0.0f), 1.0f);
        s *= 0.34f;                                   // 1.7 * mean over 5
        float shaded = (180.0f / 255.0f) * bsum * s;  // same for all channels
        float alpha = (!mask && (tz < 0.15f)) ? 1.0f : 0.0f;
        long bb  = p / HW;
        long rem = p - bb * HW;
        const float* img = images + (size_t)bb * 3 * HW + rem;
        float*       o   = out    + (size_t)bb * 3 * HW + rem;
        __builtin_prefetch(img, 0, 0);                // global_prefetch_b8
        float om = 1.0f - alpha;
        float sa = shaded * alpha;
        o[0 * HW] = sa + img[0 * HW] * om;
        o[1 * HW] = sa + img[1 * HW] * om;
        o[2 * HW] = sa + img[2 * HW] * om;
    }
}

// ---------------------------------------------------------------------------
// Host launcher
// ---------------------------------------------------------------------------
extern "C" void kernel_launch(void* const* d_in, const int* in_sizes, int n_in,
                              void* d_out, int out_size, void* d_ws, size_t ws_size,
                              hipStream_t stream) {
    (void)in_sizes; (void)n_in; (void)out_size; (void)ws_size;
    const float* vw   = (const float*)d_in[0];  // vertices_world
    const float* vi   = (const float*)d_in[1];  // vertices_image_normalised
    const int*   tf   = (const int*)  d_in[2];  // tri_faces
    const int*   p2f  = (const int*)  d_in[3];  // pix_to_face
    const float* bary = (const float*)d_in[4];  // bary_coords
    const float* imgs = (const float*)d_in[5];  // images
    float* out = (float*)d_out;

    float*  accW   = (float*)d_ws;                       // BS*NV*3 floats
    float*  accT   = accW + (size_t)BS * NV * 3;         // BS*NV*3 floats
    float4* packed = (float4*)(accT + (size_t)BS * NV * 3); // BS*NV float4

    const int nacc = 2 * BS * NV * 3;
    mr_zero_kernel<<<(nacc + 255) / 256, 256, 0, stream>>>(accW, nacc);

    mr_face_kernel<<<(BS * NTRI + 255) / 256, 256, 0, stream>>>(
        vw, vi, tf, accW, accT);

    mr_normalize_kernel<<<(BS * NV + 255) / 256, 256, 0, stream>>>(
        accW, accT, packed);

    // NPIX / 16 pixel-groups, 8 waves (groups) per 256-thread block
    mr_pixel_kernel<<<(NPIX / 16) / 8, 256, 0, stream>>>(
        p2f, bary, tf, packed, imgs, out);
}